// DTWPositionalEncoding_68650757259365
// MI455X (gfx1250) — compile-verified
//
#include <hip/hip_runtime.h>

#define B_ 64
#define S_ 512
#define T_ 512
#define D_ 128
#define INF_F __builtin_huge_valf()

typedef __attribute__((ext_vector_type(2))) float v2f;
typedef __attribute__((ext_vector_type(4))) float v4f;
typedef __attribute__((ext_vector_type(8))) float v8f;

// ---------------------------------------------------------------------------
// Kernel 0: squared row norms.  r < S_ -> source_pe row r ; else x row (r-S_).
// ---------------------------------------------------------------------------
__global__ void rownorm_kernel(const float* __restrict__ x,
                               const float* __restrict__ spe,
                               float* __restrict__ psq,
                               float* __restrict__ ssq) {
  int r = blockIdx.x * blockDim.x + threadIdx.x;
  const int nrows = B_ * T_ + S_;
  if (r >= nrows) return;
  const float* row = (r < S_) ? (spe + (size_t)r * D_)
                              : (x + (size_t)(r - S_) * D_);
  float acc = 0.f;
#pragma unroll
  for (int k = 0; k < D_; k += 4) {
    v4f v = *(const v4f*)(row + k);
    acc += v.x * v.x + v.y * v.y + v.z * v.z + v.w * v.w;
  }
  if (r < S_) ssq[r] = acc;
  else        psq[r - S_] = acc;
}

// ---------------------------------------------------------------------------
// Kernel 1: cost[b][s][t] = max(ssq[s] + psq[b][t] - 2*dot(spe[s], x[b][t]), 0)
// One wave per 16x16 tile; fp32 WMMA 16x16x4, K=128 -> 32 chained wmma ops.
// A layout (ISA 7.12.2, 32-bit A 16x4): lane<16 holds K={k,k+1} of row M=lane,
// lane>=16 holds K={k+2,k+3} of row M=lane-16.  B mirrored (K x N = 4x16).
// ---------------------------------------------------------------------------
__global__ void cost_wmma_kernel(const float* __restrict__ x,
                                 const float* __restrict__ spe,
                                 const float* __restrict__ psq,
                                 const float* __restrict__ ssq,
                                 float* __restrict__ cost) {
  const int lane = threadIdx.x & 31;
  const int wave = threadIdx.x >> 5;
  const int tile = blockIdx.x * 8 + wave;       // B * 32 * 32 = 65536 tiles
  const int b  = tile >> 10;
  const int st = (tile >> 5) & 31;
  const int tt = tile & 31;
  const int s0 = st * 16, t0 = tt * 16;
  const int m  = lane & 15;
  const int hi = lane >> 4;

  const float* arow = spe + (size_t)(s0 + m) * D_;
  const float* brow = x + ((size_t)b * T_ + (t0 + m)) * D_;

  v8f acc = {};
#pragma unroll 4
  for (int k = 0; k < D_; k += 4) {
    const int kb = k + 2 * hi;
    v2f av = *(const v2f*)(arow + kb);
    v2f bv = *(const v2f*)(brow + kb);
    // 8 args: (neg_a, A, neg_b, B, c_mod, C, reuse_a, reuse_b)
    acc = __builtin_amdgcn_wmma_f32_16x16x4_f32(false, av, false, bv,
                                                (short)0, acc, false, false);
  }

  // C/D layout: VGPR r -> M = r + 8*hi, N = m (lane).
  const float myps = psq[(size_t)b * T_ + (t0 + m)];
  size_t base = ((size_t)b * S_ + (size_t)(s0 + 8 * hi)) * T_ + (t0 + m);
#pragma unroll
  for (int r = 0; r < 8; ++r) {
    float c = ssq[s0 + 8 * hi + r] + myps - 2.0f * acc[r];
    cost[base + (size_t)r * T_] = fmaxf(c, 0.0f);
  }
}

// ---------------------------------------------------------------------------
// Kernel 2: DTW forward accumulation, anti-diagonal wavefront.
// One workgroup per sample; 544 threads (17 waves) cover rows i = 0..512 of
// the padded (S+1)x(T+1) matrix. Three rotating diagonals live in LDS.
// Emits per-cell argmin choice (0=diag,1=up,2=left, first-min tie-break,
// identical to np.argmin([diag, up, left])).
// ---------------------------------------------------------------------------
__global__ void dtw_forward_kernel(const float* __restrict__ cost,
                                   unsigned char* __restrict__ choice) {
  const int b = blockIdx.x;
  const int tid = threadIdx.x;          // 0..543
  __shared__ float buf[3][S_ + 1];

  if (tid == 0) {
    buf[0][0] = 0.0f;    // D[0][0]     (diagonal k=0)
    buf[1][0] = INF_F;   // D[0][1]     (diagonal k=1)
    buf[1][1] = INF_F;   // D[1][0]
  }
  __syncthreads();

  const float* cb = cost + (size_t)b * S_ * T_;
  unsigned char* chb = choice + (size_t)b * S_ * T_;

  for (int k = 2; k <= S_ + T_; ++k) {
    float* cur       = buf[k % 3];
    const float* p1  = buf[(k + 2) % 3];   // diagonal k-1
    const float* p2  = buf[(k + 1) % 3];   // diagonal k-2
    const int i = tid;
    const int j = k - i;
    if (i <= S_ && j >= 0 && j <= T_) {
      if (i == 0) {
        cur[0] = INF_F;                    // D[0][j], j >= 2
      } else if (j == 0) {
        cur[i] = INF_F;                    // D[i][0], i >= 1
      } else {
        const float dg   = p2[i - 1];      // D[i-1][j-1]
        const float up   = p1[i - 1];      // D[i-1][j]
        const float left = p1[i];          // D[i][j-1]
        const float d = cb[(size_t)(i - 1) * T_ + (j - 1)];
        float mn; int c;
        if (dg <= up && dg <= left) { mn = dg;   c = 0; }
        else if (up <= left)        { mn = up;   c = 1; }
        else                        { mn = left; c = 2; }
        cur[i] = d + mn;
        chb[(size_t)(i - 1) * T_ + (j - 1)] = (unsigned char)c;
      }
    }
    __syncthreads();
  }
}

// ---------------------------------------------------------------------------
// Kernel 3: backtrace (first visit per column == max source index, matching
// segment_max of the monotone path) + gather source_pe rows into output.
// ---------------------------------------------------------------------------
__global__ void backtrace_gather_kernel(const unsigned char* __restrict__ choice,
                                        const float* __restrict__ spe,
                                        float* __restrict__ out) {
  const int b = blockIdx.x;
  __shared__ int idx[T_];

  if (threadIdx.x == 0) {
    const unsigned char* chb = choice + (size_t)b * S_ * T_;
    int i = S_, j = T_, lastj = -1;
    while (i > 0 && j > 0) {
      if (j != lastj) { idx[j - 1] = i - 1; lastj = j; }
      unsigned char c = chb[(size_t)(i - 1) * T_ + (j - 1)];
      if (c == 0)      { --i; --j; }
      else if (c == 1) { --i; }
      else             { --j; }
    }
  }
  __syncthreads();

  float* ob = out + (size_t)b * T_ * D_;
  for (int n = threadIdx.x; n < T_ * (D_ / 4); n += blockDim.x) {
    const int t  = n / (D_ / 4);
    const int c4 = (n % (D_ / 4)) * 4;
    v4f v = *(const v4f*)(spe + (size_t)idx[t] * D_ + c4);
    *(v4f*)(ob + (size_t)t * D_ + c4) = v;
  }
}

// ---------------------------------------------------------------------------
extern "C" void kernel_launch(void* const* d_in, const int* in_sizes, int n_in,
                              void* d_out, int out_size, void* d_ws, size_t ws_size,
                              hipStream_t stream) {
  const float* x   = (const float*)d_in[0];   // (B, T, D) fp32
  const float* spe = (const float*)d_in[1];   // (S, D)    fp32
  float* out = (float*)d_out;                 // (B, T, D) fp32

  char* ws = (char*)d_ws;
  const size_t cost_bytes = (size_t)B_ * S_ * T_ * sizeof(float);   // 64 MB
  const size_t psq_bytes  = (size_t)B_ * T_ * sizeof(float);
  const size_t ssq_bytes  = (size_t)S_ * sizeof(float);
  float* cost = (float*)ws;
  float* psq  = (float*)(ws + cost_bytes);
  float* ssq  = (float*)(ws + cost_bytes + psq_bytes);
  unsigned char* choice = (unsigned char*)(ws + cost_bytes + psq_bytes + ssq_bytes);

  const int nrows = B_ * T_ + S_;
  rownorm_kernel<<<(nrows + 255) / 256, 256, 0, stream>>>(x, spe, psq, ssq);

  const int ntiles = B_ * (S_ / 16) * (T_ / 16);   // 65536
  cost_wmma_kernel<<<ntiles / 8, 256, 0, stream>>>(x, spe, psq, ssq, cost);

  dtw_forward_kernel<<<B_, 544, 0, stream>>>(cost, choice);

  backtrace_gather_kernel<<<B_, 256, 0, stream>>>(choice, spe, out);
}